// DeltaRuleModel_35270271435155
// MI455X (gfx1250) — compile-verified
//
#include <hip/hip_runtime.h>
#include <hip/hip_bf16.h>

// ---------------------------------------------------------------------------
// Fused DeltaRule model for MI455X (gfx1250, wave32).
//   grid  = B (1024) blocks, one batch per block
//   block = 64 threads = 2 waves:
//     wave 0: encoder (embed + MLP via V_WMMA_F32_16X16X4_F32 + residual + LN)
//     wave 1: sequential delta-rule scan (rank-1 VALU updates) + readout
//   Waves pipeline through a double-buffered 16-token LDS chunk.
//   w1/w2 are staged in LDS pre-swizzled into WMMA B-fragment order so each
//   fragment is a single conflict-free ds_load_b64 per lane.
// ---------------------------------------------------------------------------

typedef float v2f __attribute__((ext_vector_type(2)));
typedef float v8f __attribute__((ext_vector_type(8)));

#define HID    32
#define HID2   64
#define VOC    64
#define BATCH  1024
#define SEQLEN 2048
#define CHUNK  16
#define NCHUNK (SEQLEN / CHUNK)
#define HSTR   34   // hidden LDS row stride (even -> 8B-aligned b64 rows)
#define GSTR   66   // relu buffer row stride

static __device__ __forceinline__ void lds_fence() {
  // cross-lane LDS RAW inside one wave: make stores visible before loads
  asm volatile("s_wait_dscnt 0" ::: "memory");
}

__global__ void __launch_bounds__(64)
delta_rule_fused(const int*   __restrict__ seq,
                 const float* __restrict__ embed,
                 const float* __restrict__ w1,
                 const float* __restrict__ b1,
                 const float* __restrict__ w2,
                 const float* __restrict__ b2,
                 const float* __restrict__ ln_g,
                 const float* __restrict__ ln_b,
                 const float* __restrict__ read_w,
                 const float* __restrict__ read_b,
                 const float* __restrict__ out_w,
                 const float* __restrict__ out_b,
                 float*       __restrict__ out)
{
  // w1 fragments: [kb(8)][nt(4)][lane(32)][v(2)]  (B-matrix layout, ISA 7.12.2)
  __shared__ float sw1f[8 * 4 * 32 * 2];
  // w2 fragments: [kb(16)][nt(2)][lane(32)][v(2)]
  __shared__ float sw2f[16 * 2 * 32 * 2];
  __shared__ float sb1[HID2];
  __shared__ float sb2[HID];
  __shared__ float sg[HID];
  __shared__ float sbn[HID];
  __shared__ float hbuf[2][CHUNK * HSTR];  // double-buffered hidden chunk
  __shared__ float gbuf[CHUNK * GSTR];     // relu intermediate (16x64)
  __shared__ float sctx[HID];
  __shared__ float srd[HID];
  __shared__ int   sidx[CHUNK];

  const int tid  = threadIdx.x;
  const int lane = tid & 31;
  const int wid  = __builtin_amdgcn_readfirstlane(tid >> 5);  // uniform -> scalar branch
  const int half = lane >> 4;          // 0 | 1
  const int l16  = lane & 15;
  const int koff = half * 2;           // A-fragment K sub-offset (ISA 7.12.2)
  const int b    = blockIdx.x;

  // ---- stage weights into LDS once, pre-swizzled to fragment order ----
  for (int i = tid; i < 8 * 4 * 64; i += 64) {
    const int frag = i >> 6;          // kb*4 + nt
    const int w    = i & 63;
    const int ln   = w >> 1;
    const int v    = w & 1;
    const int kb   = frag >> 2, nt = frag & 3;
    const int K    = 4 * kb + ((ln >> 4) << 1) + v;
    const int N    = nt * 16 + (ln & 15);
    sw1f[i] = w1[K * HID2 + N];
  }
  for (int i = tid; i < 16 * 2 * 64; i += 64) {
    const int frag = i >> 6;          // kb*2 + nt
    const int w    = i & 63;
    const int ln   = w >> 1;
    const int v    = w & 1;
    const int kb   = frag >> 1, nt = frag & 1;
    const int K    = 4 * kb + ((ln >> 4) << 1) + v;
    const int N    = nt * 16 + (ln & 15);
    sw2f[i] = w2[K * HID + N];
  }
  if (tid < HID2) sb1[tid] = b1[tid];
  if (tid < HID) { sb2[tid] = b2[tid]; sg[tid] = ln_g[tid]; sbn[tid] = ln_b[tid]; }
  __syncthreads();

  // scanner state: lane i holds row i of the 32x32 memory M
  float m[HID];
#pragma unroll
  for (int j = 0; j < HID; ++j) m[j] = 0.f;

  // =========================== pipeline ==================================
  for (int c = 0; c <= NCHUNK; ++c) {
    if (wid == 0) {
      // ---------------- producer: encode chunk c --------------------------
      if (c < NCHUNK) {
        float* hb = hbuf[c & 1];
        const int t0 = c * CHUNK;

        if (lane < CHUNK) sidx[lane] = seq[b * SEQLEN + t0 + lane];
        lds_fence();

        // gather embedding rows: lane -> (token l16, half of the 32 dims)
        {
          const float* er  = embed + sidx[l16] * HID + half * 16;
          float*       dst = hb + l16 * HSTR + half * 16;
#pragma unroll
          for (int q = 0; q < 16; ++q) dst[q] = er[q];
        }
        lds_fence();

        // A fragments of h (16 tokens x 32), f32 16x16x4 layout
        v2f a[8];
#pragma unroll
        for (int kb = 0; kb < 8; ++kb)
          a[kb] = *(const v2f*)(hb + l16 * HSTR + kb * 4 + koff);

        // ---- GEMM1: (16x32) @ (32x64), +b1, ReLU -> gbuf ----
#pragma unroll
        for (int nt = 0; nt < 4; ++nt) {
          const int col = nt * 16 + l16;
          v8f acc = {};                              // inline-0 C on first WMMA
#pragma unroll
          for (int kb = 0; kb < 8; ++kb) {
            const v2f bf = *(const v2f*)(sw1f + ((kb * 4 + nt) * 32 + lane) * 2);
            acc = __builtin_amdgcn_wmma_f32_16x16x4_f32(
                false, a[kb], false, bf, (short)0, acc, false, false);
          }
          const float bias = sb1[col];
#pragma unroll
          for (int v = 0; v < 8; ++v) {
            float g = acc[v] + bias;
            g = g > 0.f ? g : 0.f;
            gbuf[(v + half * 8) * GSTR + col] = g;
          }
        }
        lds_fence();

        // A fragments of relu(G) (16 tokens x 64)
        v2f a2[16];
#pragma unroll
        for (int kb = 0; kb < 16; ++kb)
          a2[kb] = *(const v2f*)(gbuf + l16 * GSTR + kb * 4 + koff);

        // ---- GEMM2: (16x64) @ (64x32), +b2, +residual -> X in hbuf ----
#pragma unroll
        for (int nt = 0; nt < 2; ++nt) {
          const int col = nt * 16 + l16;
          v8f acc = {};
#pragma unroll
          for (int kb = 0; kb < 16; ++kb) {
            const v2f bf = *(const v2f*)(sw2f + ((kb * 2 + nt) * 32 + lane) * 2);
            acc = __builtin_amdgcn_wmma_f32_16x16x4_f32(
                false, a2[kb], false, bf, (short)0, acc, false, false);
          }
          const float bias = sb2[col];
#pragma unroll
          for (int v = 0; v < 8; ++v) {
            const int row = v + half * 8;
            float x = acc[v] + bias + hb[row * HSTR + col];   // residual h + ff
            hb[row * HSTR + col] = x;
          }
        }
        lds_fence();

        // ---- LayerNorm per token (lanes 0..15, one token each) ----
        if (lane < CHUNK) {
          float x[HID];
          float s = 0.f;
#pragma unroll
          for (int k = 0; k < HID; ++k) { x[k] = hb[lane * HSTR + k]; s += x[k]; }
          const float mu = s * (1.f / HID);
          float var = 0.f;
#pragma unroll
          for (int k = 0; k < HID; ++k) { float d = x[k] - mu; var += d * d; }
          const float inv = rsqrtf(var * (1.f / HID) + 1e-5f);
#pragma unroll
          for (int k = 0; k < HID; ++k)
            hb[lane * HSTR + k] = (x[k] - mu) * inv * sg[k] + sbn[k];
        }
        lds_fence();
      }
    } else {
      // ---------------- consumer: scan chunk c-1 --------------------------
      if (c >= 1) {
        const int cc = c - 1;
        const float* hb = hbuf[cc & 1];
        const int nst = (cc == NCHUNK - 1) ? (CHUNK - 1) : CHUNK;  // last token = query
        for (int t = 0; t < nst; ++t) {
          const float* kr = hb + t * HSTR;
          float kv[HID];
#pragma unroll
          for (int j = 0; j < HID / 2; ++j) {
            v2f p = *(const v2f*)(kr + 2 * j);      // broadcast b64 loads
            kv[2 * j] = p.x; kv[2 * j + 1] = p.y;
          }
          float denom = 1e-6f;
          float vp = 0.f;
#pragma unroll
          for (int j = 0; j < HID; ++j) { denom += kv[j] * kv[j]; vp += m[j] * kv[j]; }
          const float klane = kr[lane];
          const float dv = klane - vp / denom;      // delta_v for row `lane`
#pragma unroll
          for (int j = 0; j < HID; ++j) m[j] += dv * kv[j];
        }
      }
    }
    __syncthreads();   // chunk c published / chunk c-1 retired
  }

  // =========================== readout (wave 1) ===========================
  if (wid == 1) {
    const float* hb = hbuf[(NCHUNK - 1) & 1];
    const float* qr = hb + (CHUNK - 1) * HSTR;      // hidden[b, L-1, :]
    float cx = 0.f;
#pragma unroll
    for (int j = 0; j < HID; ++j) cx += m[j] * qr[j];
    sctx[lane] = cx;
    lds_fence();

    float r = read_b[lane];
#pragma unroll
    for (int i = 0; i < HID; ++i) r += sctx[i] * read_w[i * HID + lane];
    srd[lane] = r;
    lds_fence();

    float o0 = out_b[lane];
    float o1 = out_b[lane + 32];
#pragma unroll
    for (int j = 0; j < HID; ++j) {
      const float rj = srd[j];
      o0 += rj * out_w[j * VOC + lane];
      o1 += rj * out_w[j * VOC + lane + 32];
    }
    out[b * VOC + lane]      = o0;
    out[b * VOC + lane + 32] = o1;
  }
}

extern "C" void kernel_launch(void* const* d_in, const int* in_sizes, int n_in,
                              void* d_out, int out_size, void* d_ws, size_t ws_size,
                              hipStream_t stream) {
  (void)in_sizes; (void)n_in; (void)out_size; (void)d_ws; (void)ws_size;
  const int*   seq    = (const int*)  d_in[0];
  const float* embed  = (const float*)d_in[1];
  const float* w1     = (const float*)d_in[2];
  const float* b1     = (const float*)d_in[3];
  const float* w2     = (const float*)d_in[4];
  const float* b2     = (const float*)d_in[5];
  const float* ln_g   = (const float*)d_in[6];
  const float* ln_b   = (const float*)d_in[7];
  const float* read_w = (const float*)d_in[8];
  const float* read_b = (const float*)d_in[9];
  const float* out_w  = (const float*)d_in[10];
  const float* out_b  = (const float*)d_in[11];
  float* out = (float*)d_out;

  dim3 grid(BATCH), block(64);
  delta_rule_fused<<<grid, block, 0, stream>>>(
      seq, embed, w1, b1, w2, b2, ln_g, ln_b, read_w, read_b, out_w, out_b, out);
}